// MambaLayer_42494406426643
// MI455X (gfx1250) — compile-verified
//
#include <hip/hip_runtime.h>
#include <hip/hip_bf16.h>

// ---------------------------------------------------------------------------
// Mamba layer forward on MI455X (gfx1250), fp32 end-to-end.
// GEMMs: V_WMMA_F32_16X16X4_F32, 4x4 register tiling (64x64/wave),
// double-buffered software-pipelined global loads.
// ---------------------------------------------------------------------------

#define D_MODEL 512
#define INNER   1024
#define DT_RANK 32
#define D_STATE 16
#define D_CONV  4
#define BATCH   4
#define SEQLEN  1024
#define MROWS   (BATCH * SEQLEN)   // 4096

typedef __attribute__((ext_vector_type(2))) float v2f;
typedef __attribute__((ext_vector_type(8))) float v8f;

__device__ __forceinline__ v8f wmma_f32_k4(v2f a, v2f b, v8f c) {
  // (neg_a, A, neg_b, B, c_mod, C, reuse_a, reuse_b)
  return __builtin_amdgcn_wmma_f32_16x16x4_f32(
      false, a, false, b, (short)0, c, false, false);
}

__device__ __forceinline__ float silu_f(float x) {
  return x / (1.0f + __expf(-x));
}

// ---------------------------------------------------------------------------
// fp32 WMMA GEMM:  C[M,N] = X[M,K(ldx)] * W[N,K]^T   (+ optional softplus(.+b))
// One wave per 64x64 output tile; 4x4 WMMA register tiles; K pipelined 2-deep.
// Requires M%64==0, N%64==0, K%8==0 (all call sites satisfy this).
//
// Fragment layouts (wave32, ISA 7.12.2):
//   A 16x4 : lane -> A[lane%16][2*(lane/16)+j], j=0,1  (one float2 per lane)
//   B 4x16 : lane -> B[2*(lane/16)+j][lane%16]         (one float2 per lane)
//   C 16x16: vgpr v -> (M = v + 8*(lane/16), N = lane%16)
// ---------------------------------------------------------------------------
template <bool SOFTPLUS>
__global__ __launch_bounds__(32)
void gemm_wmma_f32_t(const float* __restrict__ X,
                     const float* __restrict__ W,
                     const float* __restrict__ bias,
                     float* __restrict__ C,
                     int M, int N, int K, int ldx) {
  const int lane = threadIdx.x & 31;
  const int r    = lane & 15;        // M-index in A frag / N-index in B frag
  const int kk   = (lane >> 4) << 1; // K sub-offset {0,2}
  const int n0   = blockIdx.x << 6;
  const int m0   = blockIdx.y << 6;

  const float* xr[4];
  const float* wr[4];
#pragma unroll
  for (int i = 0; i < 4; ++i) {
    xr[i] = X + (size_t)(m0 + 16 * i + r) * ldx + kk;
    wr[i] = W + (size_t)(n0 + 16 * i + r) * K   + kk;
    __builtin_prefetch(xr[i], 0, 0);   // gfx1250 global_prefetch into GL2
    __builtin_prefetch(wr[i], 0, 0);
  }

  v8f acc[4][4];
#pragma unroll
  for (int i = 0; i < 4; ++i)
#pragma unroll
    for (int j = 0; j < 4; ++j) acc[i][j] = (v8f){};

  v2f a0[4], b0[4], a1[4], b1[4];

  auto LD = [&](int k, v2f* A, v2f* Bv) {
#pragma unroll
    for (int i = 0; i < 4; ++i) {
      A[i]  = *(const v2f*)(xr[i] + k);
      Bv[i] = *(const v2f*)(wr[i] + k);
    }
  };
  auto MMA = [&](const v2f* A, const v2f* Bv) {
#pragma unroll
    for (int i = 0; i < 4; ++i)
#pragma unroll
      for (int j = 0; j < 4; ++j)
        acc[i][j] = wmma_f32_k4(A[i], Bv[j], acc[i][j]);
  };

  // 2-deep software pipeline over k-slabs of 4
  LD(0, a0, b0);
  int k = 0;
  for (; k + 8 < K; k += 8) {
    LD(k + 4, a1, b1);   // issue next slab before consuming current
    MMA(a0, b0);
    LD(k + 8, a0, b0);
    MMA(a1, b1);
  }
  LD(k + 4, a1, b1);     // k == K-8 here: final pair, no overfetch
  MMA(a0, b0);
  MMA(a1, b1);

  // Epilogue + store
  const int mloc = (lane >> 4) << 3;   // 0 or 8
  const int nloc = r;
#pragma unroll
  for (int i = 0; i < 4; ++i) {
    const int mrow = m0 + 16 * i + mloc;
#pragma unroll
    for (int j = 0; j < 4; ++j) {
      const int ncol = n0 + 16 * j + nloc;
      const float bia = SOFTPLUS ? bias[ncol] : 0.0f;
#pragma unroll
      for (int v = 0; v < 8; ++v) {
        float s = acc[i][j][v] + bia;
        if (SOFTPLUS) s = (s > 20.f) ? s : log1pf(__expf(s));
        C[(size_t)(mrow + v) * N + ncol] = s;
      }
    }
  }
}

// ---------------------------------------------------------------------------
// Depthwise causal conv1d (k=4) + bias + SiLU.
// xz is [M, 2*INNER]; xi channel d lives at column d. Output xc [M, INNER].
// ---------------------------------------------------------------------------
__global__ void conv_silu(const float* __restrict__ xz,
                          const float* __restrict__ cw,
                          const float* __restrict__ cb,
                          float* __restrict__ xc) {
  const int idx = blockIdx.x * blockDim.x + threadIdx.x; // [0, M*INNER)
  const int d = idx & (INNER - 1);
  const int m = idx >> 10;            // INNER == 1024
  const int l = m & (SEQLEN - 1);     // position within batch
  float acc = cb[d];
#pragma unroll
  for (int j = 0; j < D_CONV; ++j) {
    const int lj = l - (D_CONV - 1) + j;
    if (lj >= 0)
      acc += xz[(size_t)(m - (D_CONV - 1) + j) * (2 * INNER) + d] * cw[d * D_CONV + j];
  }
  xc[idx] = silu_f(acc);
}

// ---------------------------------------------------------------------------
// Selective scan. One lane per (batch b, channel d): 16-state recurrence in
// VGPRs, A row precomputed, B/C vectors staged through LDS in 8-step chunks.
//   h[n] = h[n]*exp(dt*A[n]) + dt*B[n]*xc ;  y = sum_n h[n]*C[n]
// ---------------------------------------------------------------------------
__global__ void ssm_scan(const float* __restrict__ proj,  // [M,64] (:,32:48)=B, (:,48:64)=C
                         const float* __restrict__ dt,    // [M, INNER]
                         const float* __restrict__ xc,    // [M, INNER]
                         const float* __restrict__ A_log, // [INNER, 16]
                         float* __restrict__ ys) {        // [M, INNER]
  const int d = blockIdx.x * 256 + threadIdx.x;
  const int b = blockIdx.y;

  float An[D_STATE];
#pragma unroll
  for (int n = 0; n < D_STATE; ++n) An[n] = -__expf(A_log[d * D_STATE + n]);

  float h[D_STATE];
#pragma unroll
  for (int n = 0; n < D_STATE; ++n) h[n] = 0.0f;

  __shared__ float bc[8][2 * D_STATE];   // [step][B(16) | C(16)]
  const int ls = threadIdx.x >> 5;       // 0..7 : step this thread loads
  const int jj = threadIdx.x & 31;       // 0..31: which of 32 B/C values

  for (int l0 = 0; l0 < SEQLEN; l0 += 8) {
    bc[ls][jj] = proj[(size_t)(b * SEQLEN + l0 + ls) * 64 + DT_RANK + jj];
    __syncthreads();
#pragma unroll 1
    for (int s = 0; s < 8; ++s) {
      const size_t m = (size_t)(b * SEQLEN + l0 + s);
      const float dtv = dt[m * INNER + d];
      const float xv  = xc[m * INNER + d];
      float y = 0.0f;
#pragma unroll
      for (int n = 0; n < D_STATE; ++n) {
        const float dA = __expf(dtv * An[n]);          // v_exp_f32
        const float hn = h[n] * dA + dtv * bc[s][n] * xv;
        h[n] = hn;
        y = fmaf(hn, bc[s][D_STATE + n], y);
      }
      ys[m * INNER + d] = y;
    }
    __syncthreads();
  }
}

// ---------------------------------------------------------------------------
// Gated combine (in-place over ys): ys = (ys + xc*D_ssm) * silu(z)
// z channel d lives at xz column INNER+d.
// ---------------------------------------------------------------------------
__global__ void combine_gate(const float* __restrict__ xcv,
                             const float* __restrict__ Dssm,
                             const float* __restrict__ xz,
                             float* __restrict__ ys_io) {
  const int idx = blockIdx.x * blockDim.x + threadIdx.x;
  const int d = idx & (INNER - 1);
  const int m = idx >> 10;
  const float zv = xz[(size_t)m * (2 * INNER) + INNER + d];
  ys_io[idx] = (ys_io[idx] + xcv[idx] * Dssm[d]) * silu_f(zv);
}

// ---------------------------------------------------------------------------
// Launch: 7 kernels on `stream`, workspace partitioned from d_ws.
// ---------------------------------------------------------------------------
extern "C" void kernel_launch(void* const* d_in, const int* in_sizes, int n_in,
                              void* d_out, int out_size, void* d_ws, size_t ws_size,
                              hipStream_t stream) {
  (void)in_sizes; (void)n_in; (void)out_size; (void)ws_size;

  const float* x      = (const float*)d_in[0]; // [B,L,D]
  const float* W_in   = (const float*)d_in[1]; // [2*INNER, D]
  const float* conv_w = (const float*)d_in[2]; // [INNER,1,4]
  const float* conv_b = (const float*)d_in[3]; // [INNER]
  const float* W_x    = (const float*)d_in[4]; // [64, INNER]
  const float* W_dt   = (const float*)d_in[5]; // [INNER, 32]
  const float* b_dt   = (const float*)d_in[6]; // [INNER]
  const float* A_log  = (const float*)d_in[7]; // [INNER, 16]
  const float* D_ssm  = (const float*)d_in[8]; // [INNER]
  const float* W_out  = (const float*)d_in[9]; // [D, INNER]
  float* out = (float*)d_out;                  // [B,L,D]

  // Workspace layout (fp32): ~81 MB total, L2-resident on the 192 MB L2.
  float* xz   = (float*)d_ws;                       // [M, 2*INNER]  32 MB
  float* xc   = xz   + (size_t)MROWS * 2 * INNER;   // [M, INNER]    16 MB
  float* proj = xc   + (size_t)MROWS * INNER;       // [M, 64]        1 MB
  float* dt   = proj + (size_t)MROWS * 64;          // [M, INNER]    16 MB
  float* ys   = dt   + (size_t)MROWS * INNER;       // [M, INNER]    16 MB (reused by combine)

  // 1) in_proj: xz = x @ W_in^T   (M=4096, N=2048, K=512)
  gemm_wmma_f32_t<false><<<dim3(2 * INNER / 64, MROWS / 64), 32, 0, stream>>>(
      x, W_in, nullptr, xz, MROWS, 2 * INNER, D_MODEL, D_MODEL);

  // 2) depthwise causal conv + SiLU -> xc
  conv_silu<<<(MROWS * INNER) / 256, 256, 0, stream>>>(xz, conv_w, conv_b, xc);

  // 3) x_proj: proj = xc @ W_x^T  (M=4096, N=64, K=1024)
  gemm_wmma_f32_t<false><<<dim3(64 / 64, MROWS / 64), 32, 0, stream>>>(
      xc, W_x, nullptr, proj, MROWS, 64, INNER, INNER);

  // 4) dt = softplus(proj[:, :32] @ W_dt^T + b_dt)  (M=4096, N=1024, K=32, ldx=64)
  gemm_wmma_f32_t<true><<<dim3(INNER / 64, MROWS / 64), 32, 0, stream>>>(
      proj, W_dt, b_dt, dt, MROWS, INNER, DT_RANK, 64);

  // 5) selective scan -> ys
  ssm_scan<<<dim3(INNER / 256, BATCH), 256, 0, stream>>>(proj, dt, xc, A_log, ys);

  // 6) gated combine in-place: ys = (ys + xc*D_ssm) * silu(z)
  combine_gate<<<(MROWS * INNER) / 256, 256, 0, stream>>>(xc, D_ssm, xz, ys);

  // 7) out_proj: out = ys @ W_out^T  (M=4096, N=512, K=1024)
  gemm_wmma_f32_t<false><<<dim3(D_MODEL / 64, MROWS / 64), 32, 0, stream>>>(
      ys, W_out, nullptr, out, MROWS, D_MODEL, INNER, INNER);
}